// GraphSAGELayer_74363063762979
// MI455X (gfx1250) — compile-verified
//
#include <hip/hip_runtime.h>

#define N_NODES 100000
#define N_EDGES 1600000
#define SZ_IN   128
#define DIM_I   1024

typedef __attribute__((ext_vector_type(2))) float v2f;
typedef __attribute__((ext_vector_type(4))) float v4f;
typedef __attribute__((ext_vector_type(8))) float v8f;

// K-interleaved fragment staging (shared by both GEMMs):
// WMMA step s=2t   consumes physical K {8t+0, 8t+1, 8t+4, 8t+5}
// WMMA step s=2t+1 consumes physical K {8t+2, 8t+3, 8t+6, 8t+7}
// A-side: one global b128 at row + 8t + 4*halfHi supplies both steps.
// B-side: lane-major LDS quad at ((nt*16 + t)*32 + lane)*4 supplies both steps:
//   quad element q holds W^T[k = 8t + 4*halfHi + q][n = nt*16 + l16].
__device__ __forceinline__ int bfrag_index(int n, int k) {
    int nt  = n >> 4, l16 = n & 15;
    int t   = k >> 3, rem = k & 7;
    int hi  = rem >> 2, q = rem & 3;
    return (((nt * 16 + t) * 32) + hi * 16 + l16) * 4 + q;
}

// ---------------------------------------------------------------------------
// Kernel 0: zero the hn scratch buffer (N_NODES x SZ_IN floats)
// ---------------------------------------------------------------------------
__global__ void zero_hn_kernel(float4* __restrict__ hn4) {
    int i = blockIdx.x * blockDim.x + threadIdx.x; // exactly 3,200,000 float4s
    hn4[i] = make_float4(0.f, 0.f, 0.f, 0.f);
}

// ---------------------------------------------------------------------------
// Kernel 1: fused edge message (relu(x[v] @ aggW^T + b)) + scatter-max to hn.
// One wave: 32 edges x 128 outputs, 16 f32-WMMA accumulators.
// Per 8-K chunk: 2 global b128 (A) + 8 ds b128 (B) feed 32 WMMAs.
// ---------------------------------------------------------------------------
#define EK_WAVES 4
__global__ __launch_bounds__(32 * EK_WAVES) void edge_msg_scatter_kernel(
    const float* __restrict__ x,        // [N_NODES, 128]
    const long long* __restrict__ ei,   // [2, N_EDGES] int64: row0=u(dst), row1=v(src)
    const float* __restrict__ aggW,     // [128, 128] row-major (out, in)
    const float* __restrict__ aggB,     // [128]
    float* __restrict__ hn)             // [N_NODES, 128]
{
    __shared__ float Wf[SZ_IN * SZ_IN]; // 64 KB, K-interleaved fragment layout

    const int tid = threadIdx.x;
    for (int i = tid; i < SZ_IN * SZ_IN; i += 32 * EK_WAVES) {
        int n = i >> 7;                 // coalesced read of aggW row-major
        int k = i & 127;
        Wf[bfrag_index(n, k)] = aggW[i];
    }
    __syncthreads();

    const int lane   = tid & 31;
    const int wave   = tid >> 5;
    const int halfHi = lane >> 4;
    const int l16    = lane & 15;

    const int tile  = blockIdx.x * EK_WAVES + wave;   // exactly 50,000 tiles
    const int eBase = tile * 32;

    const long long vsrc0 = ei[(size_t)N_EDGES + eBase + l16];
    const long long vsrc1 = ei[(size_t)N_EDGES + eBase + 16 + l16];
    const int       uDst0 = (int)ei[eBase + l16];
    const int       uDst1 = (int)ei[eBase + 16 + l16];
    const float* xrow0 = x + (size_t)vsrc0 * SZ_IN + 4 * halfHi;
    const float* xrow1 = x + (size_t)vsrc1 * SZ_IN + 4 * halfHi;
    const float* bbase = Wf + (size_t)lane * 4;

    v8f acc[16] = {};  // [0..7]: edges 0-15, [8..15]: edges 16-31

    for (int t = 0; t < SZ_IN / 8; ++t) {
        v4f a0 = *(const v4f*)(xrow0 + 8 * t);   // steps 2t (x,y) and 2t+1 (z,w)
        v4f a1 = *(const v4f*)(xrow1 + 8 * t);
        v2f a0e = __builtin_shufflevector(a0, a0, 0, 1);
        v2f a0o = __builtin_shufflevector(a0, a0, 2, 3);
        v2f a1e = __builtin_shufflevector(a1, a1, 0, 1);
        v2f a1o = __builtin_shufflevector(a1, a1, 2, 3);
        #pragma unroll
        for (int nt = 0; nt < 8; ++nt) {
            v4f bq = *(const v4f*)(bbase + (size_t)(nt * 16 + t) * 128);
            v2f be = __builtin_shufflevector(bq, bq, 0, 1);
            v2f bo = __builtin_shufflevector(bq, bq, 2, 3);
            acc[nt] = __builtin_amdgcn_wmma_f32_16x16x4_f32(
                false, a0e, false, be, (short)0, acc[nt], false, false);
            acc[nt] = __builtin_amdgcn_wmma_f32_16x16x4_f32(
                false, a0o, false, bo, (short)0, acc[nt], false, false);
            acc[8 + nt] = __builtin_amdgcn_wmma_f32_16x16x4_f32(
                false, a1e, false, be, (short)0, acc[8 + nt], false, false);
            acc[8 + nt] = __builtin_amdgcn_wmma_f32_16x16x4_f32(
                false, a1o, false, bo, (short)0, acc[8 + nt], false, false);
        }
    }

    // Epilogue: bias + relu, scatter-max into hn[u].
    float biasv[8];
    #pragma unroll
    for (int nt = 0; nt < 8; ++nt) biasv[nt] = aggB[nt * 16 + l16];

    #pragma unroll
    for (int r = 0; r < 8; ++r) {
        const int m  = r + 8 * halfHi;
        const int u0 = __shfl(uDst0, m, 32);
        const int u1 = __shfl(uDst1, m, 32);
        float* row0 = hn + (size_t)u0 * SZ_IN + l16;
        float* row1 = hn + (size_t)u1 * SZ_IN + l16;
        #pragma unroll
        for (int nt = 0; nt < 8; ++nt) {
            float v0 = fmaxf(acc[nt][r] + biasv[nt], 0.0f);
            float v1 = fmaxf(acc[8 + nt][r] + biasv[nt], 0.0f);
            if (v0 > 0.0f) atomicMax((int*)(row0 + nt * 16), __float_as_int(v0));
            if (v1 > 0.0f) atomicMax((int*)(row1 + nt * 16), __float_as_int(v1));
        }
    }
}

// ---------------------------------------------------------------------------
// Kernel 2: h = relu(concat(x, hn) @ linW^T).  M=100000, N=1024, K=256.
// grid.y picks a 128-col chunk; each K-half (x then hn) staged in 64 KB LDS
// in the K-interleaved fragment layout. One wave: 32 rows x 128 cols.
// ---------------------------------------------------------------------------
#define OK_WAVES 5
__global__ __launch_bounds__(32 * OK_WAVES) void out_gemm_kernel(
    const float* __restrict__ x,     // [N_NODES, 128]
    const float* __restrict__ hn,    // [N_NODES, 128]
    const float* __restrict__ linW,  // [1024, 256] row-major
    float* __restrict__ out)         // [N_NODES, 1024]
{
    __shared__ float Wf[128 * 128];  // 64 KB, fragment layout, one K-half

    const int tid    = threadIdx.x;
    const int nBase  = blockIdx.y * 128;
    const int lane   = tid & 31;
    const int wave   = tid >> 5;
    const int halfHi = lane >> 4;
    const int l16    = lane & 15;

    const int rowTile = blockIdx.x * OK_WAVES + wave;  // exactly 3,125 tiles
    const int row0    = rowTile * 32 + l16;
    const int row1    = row0 + 16;
    const float* xr0 = x  + (size_t)row0 * SZ_IN + 4 * halfHi;
    const float* xr1 = x  + (size_t)row1 * SZ_IN + 4 * halfHi;
    const float* hr0 = hn + (size_t)row0 * SZ_IN + 4 * halfHi;
    const float* hr1 = hn + (size_t)row1 * SZ_IN + 4 * halfHi;
    const float* bbase = Wf + (size_t)lane * 4;

    v8f acc[16] = {};

    for (int kh = 0; kh < 2; ++kh) {                   // K-half: 0 -> x, 1 -> hn
        __syncthreads();                               // protect previous stage
        for (int i = tid; i < 128 * 128; i += 32 * OK_WAVES) {
            int n = i >> 7;
            int k = i & 127;                           // coalesced along linW row
            Wf[bfrag_index(n, k)] = linW[(size_t)(nBase + n) * 256 + kh * 128 + k];
        }
        __syncthreads();

        const float* a0p = (kh == 0) ? xr0 : hr0;
        const float* a1p = (kh == 0) ? xr1 : hr1;
        for (int t = 0; t < 16; ++t) {
            v4f a0 = *(const v4f*)(a0p + 8 * t);
            v4f a1 = *(const v4f*)(a1p + 8 * t);
            v2f a0e = __builtin_shufflevector(a0, a0, 0, 1);
            v2f a0o = __builtin_shufflevector(a0, a0, 2, 3);
            v2f a1e = __builtin_shufflevector(a1, a1, 0, 1);
            v2f a1o = __builtin_shufflevector(a1, a1, 2, 3);
            #pragma unroll
            for (int nt = 0; nt < 8; ++nt) {
                v4f bq = *(const v4f*)(bbase + (size_t)(nt * 16 + t) * 128);
                v2f be = __builtin_shufflevector(bq, bq, 0, 1);
                v2f bo = __builtin_shufflevector(bq, bq, 2, 3);
                acc[nt] = __builtin_amdgcn_wmma_f32_16x16x4_f32(
                    false, a0e, false, be, (short)0, acc[nt], false, false);
                acc[nt] = __builtin_amdgcn_wmma_f32_16x16x4_f32(
                    false, a0o, false, bo, (short)0, acc[nt], false, false);
                acc[8 + nt] = __builtin_amdgcn_wmma_f32_16x16x4_f32(
                    false, a1e, false, be, (short)0, acc[8 + nt], false, false);
                acc[8 + nt] = __builtin_amdgcn_wmma_f32_16x16x4_f32(
                    false, a1o, false, bo, (short)0, acc[8 + nt], false, false);
            }
        }
    }

    // ReLU + store.
    #pragma unroll
    for (int r = 0; r < 8; ++r) {
        const int m  = r + 8 * halfHi;
        const int g0 = rowTile * 32 + m;
        const int g1 = g0 + 16;
        #pragma unroll
        for (int nt = 0; nt < 8; ++nt) {
            out[(size_t)g0 * DIM_I + nBase + nt * 16 + l16] =
                fmaxf(acc[nt][r], 0.0f);
            out[(size_t)g1 * DIM_I + nBase + nt * 16 + l16] =
                fmaxf(acc[8 + nt][r], 0.0f);
        }
    }
}

// ---------------------------------------------------------------------------
extern "C" void kernel_launch(void* const* d_in, const int* in_sizes, int n_in,
                              void* d_out, int out_size, void* d_ws, size_t ws_size,
                              hipStream_t stream) {
    const float*     x    = (const float*)d_in[0];
    const long long* ei   = (const long long*)d_in[1]; // int64 edge_index [2, E]
    const float*     aggW = (const float*)d_in[2];
    const float*     aggB = (const float*)d_in[3];
    const float*     linW = (const float*)d_in[4];
    float* out = (float*)d_out;
    float* hn  = (float*)d_ws;  // N_NODES * 128 floats = 51.2 MB scratch

    (void)in_sizes; (void)n_in; (void)out_size; (void)ws_size;

    // 1) hn = 0
    zero_hn_kernel<<<12500, 256, 0, stream>>>((float4*)hn);

    // 2) fused edge GEMM + scatter-max  (50,000 tiles = 12,500 x 4 waves)
    edge_msg_scatter_kernel<<<12500, 32 * EK_WAVES, 0, stream>>>(
        x, ei, aggW, aggB, hn);

    // 3) output GEMM + ReLU  (3,125 row tiles = 625 x 5 waves; 8 col chunks)
    out_gemm_kernel<<<dim3(625, 8), 32 * OK_WAVES, 0, stream>>>(
        x, hn, linW, out);
}